// KeywordHGNN_62749472195283
// MI455X (gfx1250) — compile-verified
//
#include <hip/hip_runtime.h>

#define N_NODES 200000
#define N_EDGES 100000
#define NNZ_CNT 2000000
#define HID 64

typedef __attribute__((ext_vector_type(2))) float v2f;
typedef __attribute__((ext_vector_type(8))) float v8f;

// ---------------------------------------------------------------------------
// Degrees: D[n] = sum_i ew[edge_i] over entries touching node n,
//          Bc[e] = |e| (hyperedge cardinality). One pass over NNZ.
// ---------------------------------------------------------------------------
__global__ __launch_bounds__(256) void hg_deg(const int* __restrict__ nidx,
                                              const int* __restrict__ eidx,
                                              const float* __restrict__ ew,
                                              float* __restrict__ D,
                                              float* __restrict__ Bc) {
  int i = blockIdx.x * 256 + threadIdx.x;
  if (i < NNZ_CNT) {
    int n = nidx[i];
    int e = eidx[i];
    unsafeAtomicAdd(&D[n], ew[e]);
    unsafeAtomicAdd(&Bc[e], 1.0f);
  }
}

// Dinv[n] = 1/D[n] (0 if empty); coef[e] = ew[e]/Bc[e]  (folds B^{-1} and w_e)
__global__ __launch_bounds__(256) void hg_inv(const float* __restrict__ D,
                                              float* __restrict__ Dinv,
                                              const float* __restrict__ Bc,
                                              const float* __restrict__ ew,
                                              float* __restrict__ coef) {
  int i = blockIdx.x * 256 + threadIdx.x;
  if (i < N_NODES) {
    float d = D[i];
    Dinv[i] = (d > 0.0f) ? 1.0f / d : 0.0f;
  }
  if (i < N_EDGES) {
    float b = Bc[i];
    coef[i] = (b > 0.0f) ? ew[i] / b : 0.0f;
  }
}

// ---------------------------------------------------------------------------
// Y = X @ W^T + b  via V_WMMA_F32_16X16X4_F32.
// One wave: 16-row block, 4 column tiles (full 64 cols), K=64 in 16 chunks.
// W (64x64 f32 = 16KB) + bias staged in LDS per workgroup (8 waves = 128 rows).
// ---------------------------------------------------------------------------
__global__ __launch_bounds__(256) void hg_gemm(const float* __restrict__ X,
                                               const float* __restrict__ W,
                                               const float* __restrict__ bias,
                                               float* __restrict__ Y,
                                               int nrows) {
  __shared__ float sW[HID * HID];
  __shared__ float sB[HID];
  for (int t = threadIdx.x; t < HID * HID; t += 256) sW[t] = W[t];
  if (threadIdx.x < HID) sB[threadIdx.x] = bias[threadIdx.x];
  __syncthreads();

  const int lane = threadIdx.x & 31;
  const int wave = threadIdx.x >> 5;
  const int rowblk = blockIdx.x * 8 + wave;
  if (rowblk * 16 >= nrows) return;          // whole-wave exit: EXEC stays all-1s

  const int m0   = rowblk * 16;
  const int mrow = m0 + (lane & 15);          // A fragment row for this lane
  const int kk   = (lane >> 4) * 2;           // K sub-pair within 4-wide chunk
  const int ncol = lane & 15;                 // B/D column within a 16-col tile

  v8f acc0 = {}, acc1 = {}, acc2 = {}, acc3 = {};
  const float* xrow = X + (size_t)mrow * HID;

  for (int kc = 0; kc < 16; ++kc) {
    const int kb = kc * 4 + kk;
    // A: 16x4 tile of X   (lane L: M = L&15, K = kb..kb+1)
    v2f a = *(const v2f*)(xrow + kb);
    // B: 4x16 tiles of W^T, i.e. B[k][n] = W[n][k]  (same per-lane K mapping)
    v2f b0 = *(const v2f*)(sW + (0 * 16 + ncol) * HID + kb);
    v2f b1 = *(const v2f*)(sW + (1 * 16 + ncol) * HID + kb);
    v2f b2 = *(const v2f*)(sW + (2 * 16 + ncol) * HID + kb);
    v2f b3 = *(const v2f*)(sW + (3 * 16 + ncol) * HID + kb);
    acc0 = __builtin_amdgcn_wmma_f32_16x16x4_f32(false, a, false, b0, (short)0, acc0, false, false);
    acc1 = __builtin_amdgcn_wmma_f32_16x16x4_f32(false, a, false, b1, (short)0, acc1, false, false);
    acc2 = __builtin_amdgcn_wmma_f32_16x16x4_f32(false, a, false, b2, (short)0, acc2, false, false);
    acc3 = __builtin_amdgcn_wmma_f32_16x16x4_f32(false, a, false, b3, (short)0, acc3, false, false);
  }

  // D layout: VGPR r -> M = r + 8*(lane>>4), N = lane&15
  const int mbase = m0 + 8 * (lane >> 4);
  for (int r = 0; r < 8; ++r) {
    const size_t row = (size_t)(mbase + r) * HID;
    Y[row + 0 * 16 + ncol] = acc0[r] + sB[0 * 16 + ncol];
    Y[row + 1 * 16 + ncol] = acc1[r] + sB[1 * 16 + ncol];
    Y[row + 2 * 16 + ncol] = acc2[r] + sB[2 * 16 + ncol];
    Y[row + 3 * 16 + ncol] = acc3[r] + sB[3 * 16 + ncol];
  }
}

// ---------------------------------------------------------------------------
// node -> hyperedge scatter: efeat[e] += T[n]   (one thread per (nnz, col))
// ---------------------------------------------------------------------------
__global__ __launch_bounds__(256) void hg_scatter_ne(const float* __restrict__ T,
                                                     const int* __restrict__ nidx,
                                                     const int* __restrict__ eidx,
                                                     float* __restrict__ efeat) {
  unsigned gid = blockIdx.x * 256u + threadIdx.x;   // NNZ*64 = 128M total
  int i = gid >> 6;
  int c = gid & 63;
  int n = nidx[i];
  int e = eidx[i];
  unsafeAtomicAdd(&efeat[(size_t)e * HID + c], T[(size_t)n * HID + c]);
}

// ---------------------------------------------------------------------------
// hyperedge -> node scatter: out[n] += coef[e] * efeat[e]
// ---------------------------------------------------------------------------
__global__ __launch_bounds__(256) void hg_scatter_en(const float* __restrict__ efeat,
                                                     const float* __restrict__ coef,
                                                     const int* __restrict__ nidx,
                                                     const int* __restrict__ eidx,
                                                     float* __restrict__ out) {
  unsigned gid = blockIdx.x * 256u + threadIdx.x;
  int i = gid >> 6;
  int c = gid & 63;
  int n = nidx[i];
  int e = eidx[i];
  unsafeAtomicAdd(&out[(size_t)n * HID + c], coef[e] * efeat[(size_t)e * HID + c]);
}

// out[n,c] *= Dinv[n]; optional ReLU
__global__ __launch_bounds__(256) void hg_scale(float* __restrict__ out,
                                                const float* __restrict__ Dinv,
                                                int relu) {
  unsigned gid = blockIdx.x * 256u + threadIdx.x;   // N_NODES*64 = 12.8M
  int n = gid >> 6;
  float v = out[gid] * Dinv[n];
  if (relu) v = fmaxf(v, 0.0f);
  out[gid] = v;
}

// ---------------------------------------------------------------------------
extern "C" void kernel_launch(void* const* d_in, const int* in_sizes, int n_in,
                              void* d_out, int out_size, void* d_ws, size_t ws_size,
                              hipStream_t stream) {
  const int*   hidx = (const int*)d_in[0];            // [2, NNZ] stacked
  const int*   nidx = hidx;                           // row 0: node ids
  const int*   eidx = hidx + NNZ_CNT;                 // row 1: edge ids
  const float* ew   = (const float*)d_in[1];
  const float* emb  = (const float*)d_in[2];
  const float* W0   = (const float*)d_in[3];
  const float* b0   = (const float*)d_in[4];
  const float* W1   = (const float*)d_in[5];
  const float* b1   = (const float*)d_in[6];
  const float* W2   = (const float*)d_in[7];
  const float* b2   = (const float*)d_in[8];
  float* outp = (float*)d_out;

  // Workspace carve-up (floats): tbuf[N*H] | bufA[N*H] | efeat[E*H] | D[N] | Bc[E] | Dinv[N] | coef[E]
  float* ws    = (float*)d_ws;
  float* tbuf  = ws;
  float* bufA  = tbuf + (size_t)N_NODES * HID;
  float* efeat = bufA + (size_t)N_NODES * HID;
  float* Dv    = efeat + (size_t)N_EDGES * HID;
  float* Bc    = Dv + N_NODES;
  float* Dinv  = Bc + N_EDGES;
  float* coef  = Dinv + N_NODES;

  // Layer-invariant degree terms
  hipMemsetAsync(Dv, 0, (size_t)(N_NODES + N_EDGES) * sizeof(float), stream); // D + Bc contiguous
  hg_deg<<<(NNZ_CNT + 255) / 256, 256, 0, stream>>>(nidx, eidx, ew, Dv, Bc);
  hg_inv<<<(N_NODES + 255) / 256, 256, 0, stream>>>(Dv, Dinv, Bc, ew, coef);

  const int gemm_grid  = (N_NODES / 16 + 7) / 8;          // 12500 row-tiles, 8 waves/block
  const int nnz_grid   = (NNZ_CNT * HID) / 256;           // 500000
  const int node_grid  = (N_NODES * HID) / 256;           // 50000

  auto layer = [&](const float* xin, const float* W, const float* b,
                   float* outb, int relu) {
    hg_gemm<<<gemm_grid, 256, 0, stream>>>(xin, W, b, tbuf, N_NODES);
    hipMemsetAsync(efeat, 0, (size_t)N_EDGES * HID * sizeof(float), stream);
    hg_scatter_ne<<<nnz_grid, 256, 0, stream>>>(tbuf, nidx, eidx, efeat);
    hipMemsetAsync(outb, 0, (size_t)N_NODES * HID * sizeof(float), stream);
    hg_scatter_en<<<nnz_grid, 256, 0, stream>>>(efeat, coef, nidx, eidx, outb);
    hg_scale<<<node_grid, 256, 0, stream>>>(outb, Dinv, relu);
  };

  layer(emb,  W0, b0, bufA, 1);   // L0: emb -> bufA (ReLU)
  layer(bufA, W1, b1, bufA, 1);   // L1: bufA consumed by GEMM first, then reused as output
  layer(bufA, W2, b2, outp, 0);   // L2: -> d_out (no ReLU)
}